// MyDecoderRNN_39204461478777
// MI455X (gfx1250) — compile-verified
//
#include <hip/hip_runtime.h>
#include <cmath>

// Shapes (fixed by the reference)
#define B_SZ    32
#define N_REG   196
#define E_DIM   512
#define D_DIM   512
#define EMB_DIM 512
#define V_SZ    32000
#define T_STEPS 64
#define CAPLEN  65

typedef __attribute__((ext_vector_type(2))) float v2f;
typedef __attribute__((ext_vector_type(8))) float v8f;

#define WMMA_F32(a, b, c) \
    __builtin_amdgcn_wmma_f32_16x16x4_f32(false, (a), false, (b), (short)0, (c), false, false)

// =====================================================================================
// Generic fp32 WMMA GEMM (small-M path):  C[M,N] = A[M,K] * Bt[N,K]^T (+bias)(+C)
// Grid: x = N/256, y = M/16.  256 threads = 8 waves; each wave: 2 N-tiles.
// =====================================================================================
#define KC           512
#define LDSA_STRIDE  (KC + 4)
#define NWAVES       8
#define TN           2

__global__ __launch_bounds__(256) void gemm_f32_wmma(
    const float* __restrict__ A, int lda,
    const float* __restrict__ Bt, int ldb,
    const float* __restrict__ bias,
    float* __restrict__ C,
    int K, int accFlag, int ldc)
{
    __shared__ float sA[16 * LDSA_STRIDE];
    const int tid  = threadIdx.x;
    const int wave = tid >> 5;
    const int lane = tid & 31;
    const int l16  = lane & 15;
    const int kOff = (lane >> 4) << 1;         // lanes 16-31 hold K+2,K+3

    const int mBase = blockIdx.y << 4;
    const int nBase = (blockIdx.x * NWAVES + wave) * (16 * TN);

    v8f acc0 = {};
    v8f acc1 = {};

    for (int k0 = 0; k0 < K; k0 += KC) {
        __syncthreads();
        for (int i = tid * 4; i < 16 * KC; i += 256 * 4) {
            const int m  = i >> 9;             // i / KC   (KC == 512)
            const int kk = i & (KC - 1);
            const float4 v = *reinterpret_cast<const float4*>(
                &A[(size_t)(mBase + m) * lda + k0 + kk]);
            *reinterpret_cast<float4*>(&sA[m * LDSA_STRIDE + kk]) = v;
        }
        __syncthreads();

        const float* b0p = &Bt[(size_t)(nBase + l16) * ldb + k0 + kOff];
        const float* b1p = b0p + (size_t)16 * ldb;
        const float* a0p = &sA[l16 * LDSA_STRIDE + kOff];

        #pragma unroll 4
        for (int kc = 0; kc < KC; kc += 4) {
            v2f afrag  = *reinterpret_cast<const v2f*>(a0p + kc);
            v2f bfrag0 = *reinterpret_cast<const v2f*>(b0p + kc);
            v2f bfrag1 = *reinterpret_cast<const v2f*>(b1p + kc);
            acc0 = WMMA_F32(afrag, bfrag0, acc0);
            acc1 = WMMA_F32(afrag, bfrag1, acc1);
        }
    }

    const int rowHalf = (lane >> 4) << 3;
    for (int tn = 0; tn < TN; ++tn) {
        const int col  = nBase + tn * 16 + l16;
        const float bv = bias ? bias[col] : 0.0f;
        v8f a = (tn == 0) ? acc0 : acc1;
        for (int j = 0; j < 8; ++j) {
            const int row = mBase + rowHalf + j;
            const size_t off = (size_t)row * ldc + col;
            float v = a[j] + bv;
            if (accFlag) v += C[off];
            C[off] = v;
        }
    }
}

// =====================================================================================
// Register-blocked fp32 WMMA GEMM (large-M path): 4 M-tiles x 2 N-tiles per wave.
// Block covers 64 rows x 256 cols. Grid: x = N/256, y = M/64.
// 8 WMMAs per 6 loads per k-step; B (weight) L2 traffic reduced 4x vs generic kernel.
// outMode==1 remaps row r = t*B+b -> out[b][t][col] (hoisted vocab projection).
// =====================================================================================
#define KC2    128
#define LDS2   (KC2 + 4)   // stride 132: lanes hit banks 4l..4l+3 -> conflict-free

__global__ __launch_bounds__(256) void gemm_f32_wmma_big(
    const float* __restrict__ A, int lda,
    const float* __restrict__ Bt, int ldb,
    const float* __restrict__ bias,
    float* __restrict__ C,
    int K, int outMode, int ldc)
{
    __shared__ float sA[64 * LDS2];
    const int tid  = threadIdx.x;
    const int wave = tid >> 5;
    const int lane = tid & 31;
    const int l16  = lane & 15;
    const int kOff = (lane >> 4) << 1;

    const int mBase = blockIdx.y << 6;                 // 64 rows per block
    const int nBase = (blockIdx.x * NWAVES + wave) * (16 * TN);

    v8f acc[8];
    #pragma unroll
    for (int i = 0; i < 8; ++i) acc[i] = (v8f){};

    for (int k0 = 0; k0 < K; k0 += KC2) {
        __syncthreads();
        // cooperative A-chunk load: 64 x 128 floats, float4 per thread per iter (8 iters)
        for (int i = tid * 4; i < 64 * KC2; i += 256 * 4) {
            const int m  = i >> 7;                     // i / KC2  (KC2 == 128)
            const int kk = i & (KC2 - 1);
            const float4 v = *reinterpret_cast<const float4*>(
                &A[(size_t)(mBase + m) * lda + k0 + kk]);
            *reinterpret_cast<float4*>(&sA[m * LDS2 + kk]) = v;
        }
        __syncthreads();

        const float* b0p = &Bt[(size_t)(nBase + l16) * ldb + k0 + kOff];
        const float* b1p = b0p + (size_t)16 * ldb;
        const float* a0p = &sA[l16 * LDS2 + kOff];

        #pragma unroll 2
        for (int kc = 0; kc < KC2; kc += 4) {
            const v2f bf0 = *reinterpret_cast<const v2f*>(b0p + kc);
            const v2f bf1 = *reinterpret_cast<const v2f*>(b1p + kc);
            #pragma unroll
            for (int mt = 0; mt < 4; ++mt) {
                const v2f af = *reinterpret_cast<const v2f*>(a0p + mt * (16 * LDS2) + kc);
                acc[mt * 2]     = WMMA_F32(af, bf0, acc[mt * 2]);
                acc[mt * 2 + 1] = WMMA_F32(af, bf1, acc[mt * 2 + 1]);
            }
        }
    }

    const int rowHalf = (lane >> 4) << 3;
    #pragma unroll
    for (int mt = 0; mt < 4; ++mt) {
        for (int tn = 0; tn < TN; ++tn) {
            const int col  = nBase + tn * 16 + l16;
            const float bv = bias ? bias[col] : 0.0f;
            v8f a = acc[mt * 2 + tn];
            for (int j = 0; j < 8; ++j) {
                const int row = mBase + mt * 16 + rowHalf + j;
                size_t off;
                if (outMode == 1) {                    // row = t*B + b -> out[b][t][col]
                    const int tt = row >> 5;           // / B_SZ (32)
                    const int bb = row & 31;
                    off = ((size_t)bb * T_STEPS + tt) * (size_t)ldc + col;
                } else {
                    off = (size_t)row * ldc + col;
                }
                C[off] = a[j] + bv;
            }
        }
    }
}

// ---------------- feat_mean over the N image regions --------------------------------
__global__ __launch_bounds__(256) void feat_mean_kernel(
    const float* __restrict__ feat, float* __restrict__ fm)
{
    const int b = blockIdx.x, tid = threadIdx.x;
    for (int col = tid; col < E_DIM; col += 256) {
        float s = 0.0f;
        for (int n = 0; n < N_REG; ++n)
            s += feat[((size_t)(b * N_REG + n)) * E_DIM + col];
        fm[b * E_DIM + col] = s * (1.0f / (float)N_REG);
    }
}

// ---------------- Per-step attention: e = v . tanh(fp + hW), softmax, ctx, emb gather
__global__ __launch_bounds__(256) void attention_step(
    const float* __restrict__ feat_proj, const float* __restrict__ hW,
    const float* __restrict__ v_att, const float* __restrict__ image_features,
    const int* __restrict__ captions, const float* __restrict__ emb,
    float* __restrict__ xbuf, float* __restrict__ attn_out, int t)
{
    __shared__ float s_hw[D_DIM];
    __shared__ float s_v[D_DIM];
    __shared__ float s_a[256];
    __shared__ float s_red[256];
    const int b = blockIdx.x, tid = threadIdx.x;

    s_hw[tid]       = hW[b * D_DIM + tid];
    s_hw[256 + tid] = hW[b * D_DIM + 256 + tid];
    s_v[tid]        = v_att[tid];
    s_v[256 + tid]  = v_att[256 + tid];
    __syncthreads();

    float e = -3.0e38f;
    if (tid < N_REG) {
        const float* fp = feat_proj + ((size_t)(b * N_REG + tid)) * D_DIM;
        float s = 0.0f;
        for (int d = 0; d < D_DIM; ++d)
            s += s_v[d] * tanhf(fp[d] + s_hw[d]);
        e = s;
    }
    s_a[tid] = e;
    s_red[tid] = e;
    __syncthreads();
    for (int s = 128; s > 0; s >>= 1) {
        if (tid < s) s_red[tid] = fmaxf(s_red[tid], s_red[tid + s]);
        __syncthreads();
    }
    const float mx = s_red[0];
    __syncthreads();
    float ex = (tid < N_REG) ? expf(s_a[tid] - mx) : 0.0f;
    s_red[tid] = ex;
    __syncthreads();
    for (int s = 128; s > 0; s >>= 1) {
        if (tid < s) s_red[tid] += s_red[tid + s];
        __syncthreads();
    }
    const float inv = 1.0f / s_red[0];
    __syncthreads();
    const float alpha = ex * inv;
    if (tid < N_REG) {
        s_a[tid] = alpha;
        attn_out[((size_t)b * T_STEPS + t) * N_REG + tid] = alpha;
    }
    __syncthreads();

    // ctx = alpha @ image_features   -> xbuf[b][512..1023]
    for (int col = tid; col < E_DIM; col += 256) {
        float s = 0.0f;
        for (int n = 0; n < N_REG; ++n)
            s += s_a[n] * image_features[((size_t)(b * N_REG + n)) * E_DIM + col];
        xbuf[b * (EMB_DIM + E_DIM) + EMB_DIM + col] = s;
    }
    // xt = emb[captions[b][t]]       -> xbuf[b][0..511]
    const int tok = captions[b * CAPLEN + t];
    for (int col = tid; col < EMB_DIM; col += 256)
        xbuf[b * (EMB_DIM + E_DIM) + col] = emb[(size_t)tok * EMB_DIM + col];
}

// ---------------- LSTM elementwise; also records h_t into Hall for the hoisted FC ---
__device__ __forceinline__ float sigmoidf_(float x) { return 1.0f / (1.0f + expf(-x)); }

__global__ __launch_bounds__(256) void lstm_ew(
    const float* __restrict__ gates, float* __restrict__ h, float* __restrict__ c,
    float* __restrict__ Hall, int t)
{
    const int idx = blockIdx.x * 256 + threadIdx.x;     // B*D = 16384
    if (idx >= B_SZ * D_DIM) return;
    const int b = idx >> 9, d = idx & (D_DIM - 1);
    const float* g = gates + (size_t)b * 4 * D_DIM;
    const float ig = sigmoidf_(g[d]);
    const float fg = sigmoidf_(g[D_DIM + d]);
    const float gg = tanhf(g[2 * D_DIM + d]);
    const float og = sigmoidf_(g[3 * D_DIM + d]);
    const float cn = fg * c[idx] + ig * gg;
    const float hn = og * tanhf(cn);
    c[idx] = cn;
    h[idx] = hn;
    Hall[(size_t)t * (B_SZ * D_DIM) + idx] = hn;        // row t*B+b
}

// ---------------- Host: stream-ordered launch sequence ------------------------------
extern "C" void kernel_launch(void* const* d_in, const int* in_sizes, int n_in,
                              void* d_out, int out_size, void* d_ws, size_t ws_size,
                              hipStream_t stream) {
    const float* imgf  = (const float*)d_in[0];
    const int*   caps  = (const int*)  d_in[1];
    const float* emb   = (const float*)d_in[2];
    const float* Wh0   = (const float*)d_in[3];
    const float* bh0   = (const float*)d_in[4];
    const float* Wc0   = (const float*)d_in[5];
    const float* bc0   = (const float*)d_in[6];
    const float* Wenc  = (const float*)d_in[7];
    const float* Wdec  = (const float*)d_in[8];
    const float* v_att = (const float*)d_in[9];
    const float* W_ih  = (const float*)d_in[10];
    const float* b_ih  = (const float*)d_in[11];
    const float* W_hh  = (const float*)d_in[12];
    const float* b_hh  = (const float*)d_in[13];
    const float* Wfc   = (const float*)d_in[14];
    const float* bfc   = (const float*)d_in[15];

    float* out      = (float*)d_out;
    float* attn_out = out + (size_t)B_SZ * T_STEPS * V_SZ;

    float* ws    = (float*)d_ws;
    float* fm    = ws;                       // 32*512
    float* h     = fm    + B_SZ * D_DIM;     // 32*512
    float* c     = h     + B_SZ * D_DIM;     // 32*512
    float* hW    = c     + B_SZ * D_DIM;     // 32*512
    float* xbuf  = hW    + B_SZ * D_DIM;     // 32*1024
    float* gates = xbuf  + B_SZ * (EMB_DIM + E_DIM);     // 32*2048
    float* fproj = gates + B_SZ * 4 * D_DIM;             // 6272*512
    float* Hall  = fproj + (size_t)B_SZ * N_REG * D_DIM; // 2048*512

    // feat_mean, then h0/c0 (WMMA), then feat_proj (register-blocked WMMA)
    feat_mean_kernel<<<B_SZ, 256, 0, stream>>>(imgf, fm);
    gemm_f32_wmma<<<dim3(2, 2), 256, 0, stream>>>(fm, E_DIM, Wh0, E_DIM, bh0, h,
                                                  E_DIM, 0, D_DIM);
    gemm_f32_wmma<<<dim3(2, 2), 256, 0, stream>>>(fm, E_DIM, Wc0, E_DIM, bc0, c,
                                                  E_DIM, 0, D_DIM);
    gemm_f32_wmma_big<<<dim3(2, (B_SZ * N_REG) / 64), 256, 0, stream>>>(
        imgf, E_DIM, Wenc, E_DIM, nullptr, fproj, E_DIM, 0, D_DIM);

    // Sequential recurrence (vocab projection hoisted out of the loop)
    for (int t = 0; t < T_STEPS; ++t) {
        gemm_f32_wmma<<<dim3(2, 2), 256, 0, stream>>>(h, D_DIM, Wdec, D_DIM, nullptr,
                                                      hW, D_DIM, 0, D_DIM);
        attention_step<<<B_SZ, 256, 0, stream>>>(fproj, hW, v_att, imgf, caps, emb,
                                                 xbuf, attn_out, t);
        gemm_f32_wmma<<<dim3(8, 2), 256, 0, stream>>>(xbuf, EMB_DIM + E_DIM,
                                                      W_ih, EMB_DIM + E_DIM, b_ih,
                                                      gates, EMB_DIM + E_DIM, 0,
                                                      4 * D_DIM);
        gemm_f32_wmma<<<dim3(8, 2), 256, 0, stream>>>(h, D_DIM, W_hh, D_DIM, b_hh,
                                                      gates, D_DIM, 1, 4 * D_DIM);
        lstm_ew<<<(B_SZ * D_DIM) / 256, 256, 0, stream>>>(gates, h, c, Hall, t);
    }

    // One big vocab GEMM: Hall[2048,512] x Wfc[32000,512]^T -> out[b][t][v]
    gemm_f32_wmma_big<<<dim3(V_SZ / 256, (T_STEPS * B_SZ) / 64), 256, 0, stream>>>(
        Hall, D_DIM, Wfc, D_DIM, bfc, out, D_DIM, 1, V_SZ);
}